// IntensityLoss_33328946217419
// MI455X (gfx1250) — compile-verified
//
#include <hip/hip_runtime.h>
#include <cstdint>
#include <cstddef>

// ---------------- problem constants ----------------
#define H_IMG 512
#define W_IMG 512
#define HW_IMG (512 * 512)
#define NBATCH 4
#define NW 25          // 5x5 window
#define TS 16          // tile size (pixels per block edge)
#define HALO 2
#define TW (TS + 2 * HALO)        // 20 (LDS tile edge)
#define NTILES (W_IMG / TS)       // 32
#define NBLOCKS_MAIN (NBATCH * NTILES * NTILES)  // 4096

// CDNA5 async global->LDS staging path (builtin confirmed present on this toolchain)
#ifndef HAVE_ASYNC_LDS
#if defined(__has_builtin)
#if __has_builtin(__builtin_amdgcn_global_load_async_to_lds_b32)
#define HAVE_ASYNC_LDS 1
#else
#define HAVE_ASYNC_LDS 0
#endif
#else
#define HAVE_ASYNC_LDS 0
#endif
#endif

#if HAVE_ASYNC_LDS
typedef __attribute__((address_space(1))) int g_i32;  // global (AS1)
typedef __attribute__((address_space(3))) int l_i32;  // LDS (AS3)
#endif

// ---------------- kernel: main stencil pass (placed first so disasm snippet shows it) ----
// grid (NTILES, NTILES, NBATCH), block (16,16). Writes per-block double partials.
__global__ void __launch_bounds__(256)
il_main(const float* __restrict__ fake,
        const float* __restrict__ gamma,
        const float* __restrict__ hdr,
        const float* __restrict__ rw,
        const float* __restrict__ ff,
        const unsigned int* __restrict__ gmax,
        double* __restrict__ part1,
        double* __restrict__ part2) {
  __shared__ float sF[TW * TW];
  __shared__ float sG[TW * TW];
  __shared__ float sH[TW * TW];
  __shared__ double red1[8];
  __shared__ double red2[8];

  const int b   = blockIdx.z;
  const int ty0 = (int)blockIdx.y * TS - HALO;
  const int tx0 = (int)blockIdx.x * TS - HALO;
  const int tid = threadIdx.y * TS + threadIdx.x;

  const float* Fb = fake  + (size_t)b * HW_IMG;
  const float* Gb = gamma + (size_t)b * HW_IMG;
  const float* Hb = hdr   + (size_t)b * HW_IMG;

  const bool interior =
      (ty0 >= 0) && (tx0 >= 0) && (ty0 + TW <= H_IMG) && (tx0 + TW <= W_IMG);
  (void)interior;

#if HAVE_ASYNC_LDS
  if (interior) {
    // CDNA5 async copy: global -> LDS without VGPR round trip (ASYNCcnt tracked)
    for (int e = tid; e < TW * TW; e += TS * TS) {
      const int ly = e / TW, lx = e % TW;
      const size_t gidx = (size_t)(ty0 + ly) * W_IMG + (size_t)(tx0 + lx);
      __builtin_amdgcn_global_load_async_to_lds_b32(
          (g_i32*)(Fb + gidx), (l_i32*)(&sF[e]), 0, 0);
      __builtin_amdgcn_global_load_async_to_lds_b32(
          (g_i32*)(Gb + gidx), (l_i32*)(&sG[e]), 0, 0);
      __builtin_amdgcn_global_load_async_to_lds_b32(
          (g_i32*)(Hb + gidx), (l_i32*)(&sH[e]), 0, 0);
    }
    asm volatile("s_wait_asynccnt 0" ::: "memory");
  } else
#endif
  {
    // border tiles: predicated scalar staging with zero padding (matches jnp.pad)
    for (int e = tid; e < TW * TW; e += TS * TS) {
      const int ly = e / TW, lx = e % TW;
      const int gy = ty0 + ly, gx = tx0 + lx;
      const bool inb = (gy >= 0) && (gy < H_IMG) && (gx >= 0) && (gx < W_IMG);
      const size_t gidx = (size_t)gy * W_IMG + (size_t)gx;
      sF[e] = inb ? Fb[gidx] : 0.0f;
      sG[e] = inb ? Gb[gidx] : 0.0f;
      sH[e] = inb ? Hb[gidx] : 0.0f;
    }
  }
  __syncthreads();

  // hdr_pow = hdr^(1-ff), applied before padding in the reference:
  // in-bounds hdr >= 0.01 so v>0 selects exactly the in-bounds entries.
  const float ffb  = ff[b];
  const float pexp = 1.0f - ffb;
  for (int e = tid; e < TW * TW; e += TS * TS) {
    const float v = sH[e];
    sH[e] = (v > 0.0f) ? powf(v, pexp) : 0.0f;
  }
  __syncthreads();

  const int x = (int)blockIdx.x * TS + threadIdx.x;
  const int y = (int)blockIdx.y * TS + threadIdx.y;

  // 25 weight planes: 105 MB streamed exactly once -> non-temporal hint so the
  // reused 16 MB of images stay resident in the 192 MB L2
  const float* RWp = rw + (size_t)b * NW * HW_IMG + (size_t)y * W_IMG + (size_t)x;
  float rwv[NW];
  float wsum = 0.0f;
#pragma unroll
  for (int k = 0; k < NW; ++k) {
    rwv[k] = __builtin_nontemporal_load(RWp + (size_t)k * HW_IMG);
    wsum += rwv[k];
  }

  // weighted sums (dist_gauss cancels in all normalized means)
  float sf1 = 0.f, sf2 = 0.f, sg1 = 0.f, sg2 = 0.f, sh1 = 0.f;
#pragma unroll
  for (int i = 0; i < 5; ++i) {
#pragma unroll
    for (int j = 0; j < 5; ++j) {
      const int e = (threadIdx.y + i) * TW + (threadIdx.x + j);
      const float wv = rwv[i * 5 + j];
      const float fv = sF[e];
      const float gv = sG[e];
      sf1 = fmaf(wv, fv, sf1);
      sf2 = fmaf(wv * fv, fv, sf2);
      sg1 = fmaf(wv, gv, sg1);
      sg2 = fmaf(wv * gv, gv, sg2);
      sh1 = fmaf(wv, sH[e], sh1);
    }
  }

  const float inv  = 1.0f / wsum;
  const float muF  = sf1 * inv;
  const float varF = fmaxf(sf2 * inv - muF * muF, 0.0f);   // == E[(x-mu)^2]
  const float stdF = sqrtf(varF + 1e-5f);
  const float muG  = sg1 * inv;
  const float varG = fmaxf(sg2 * inv - muG * muG, 0.0f);
  const float stdG = sqrtf(varG + 1e-5f);
  const float hpmu = sh1 * inv;

  const float gm     = __uint_as_float(gmax[b]);
  const float stdObj = (1.0f / ffb) * stdG * (hpmu + 1e-5f) * gm;  // ALPHA = 1
  const float base   = 1.0f - stdF / (stdF + stdObj);
  const float wblf   = wsum - 1.0f;

  // mean(res) == sum(base*w_blf) / sum(w_blf)  (the mean(w_blf) normalization cancels)
  double c1 = (double)(base * wblf);
  double c2 = (double)wblf;
#pragma unroll
  for (int o = 16; o > 0; o >>= 1) {
    c1 += __shfl_down(c1, o);
    c2 += __shfl_down(c2, o);
  }
  if ((tid & 31) == 0) { red1[tid >> 5] = c1; red2[tid >> 5] = c2; }
  __syncthreads();
  if (tid == 0) {
    double t1 = 0.0, t2 = 0.0;
#pragma unroll
    for (int i = 0; i < 8; ++i) { t1 += red1[i]; t2 += red2[i]; }
    const int bid = ((int)blockIdx.z * (int)gridDim.y + (int)blockIdx.y) * (int)gridDim.x +
                    (int)blockIdx.x;
    part1[bid] = t1;
    part2[bid] = t2;
  }
}

// ---------------- kernel: init workspace (gray-max slots) ----------------
__global__ void il_init_ws(unsigned int* __restrict__ gmax) {
  if (threadIdx.x < NBATCH) gmax[threadIdx.x] = 0u;   // floats are >=0, bits 0 == 0.0f
}

// ---------------- kernel: per-batch max of hdr_original_gray ----------------
// grid (256, NBATCH), block 256; one float4 per thread -> 256*256*4 = 262144 = HW
__global__ void __launch_bounds__(256)
il_gray_max(const float* __restrict__ gray, unsigned int* __restrict__ gmax) {
  const int b = blockIdx.y;
  const float4* g = (const float4*)(gray + (size_t)b * HW_IMG + (size_t)blockIdx.x * 1024) +
                    threadIdx.x;
  const float4 v = *g;   // GLOBAL_LOAD_B128
  float m = fmaxf(fmaxf(v.x, v.y), fmaxf(v.z, v.w));
#pragma unroll
  for (int o = 16; o > 0; o >>= 1) m = fmaxf(m, __shfl_down(m, o));  // wave32
  __shared__ float wmax[8];
  const int tid = threadIdx.x;
  if ((tid & 31) == 0) wmax[tid >> 5] = m;
  __syncthreads();
  if (tid == 0) {
    float mm = wmax[0];
#pragma unroll
    for (int i = 1; i < 8; ++i) mm = fmaxf(mm, wmax[i]);
    // exact & order-independent for non-negative floats -> deterministic
    atomicMax(&gmax[b], __float_as_uint(mm));
  }
}

// ---------------- kernel: deterministic final reduce ----------------
__global__ void __launch_bounds__(256)
il_finalize(const double* __restrict__ part1, const double* __restrict__ part2,
            float* __restrict__ out, int n) {
  double s1 = 0.0, s2 = 0.0;
  for (int i = threadIdx.x; i < n; i += 256) { s1 += part1[i]; s2 += part2[i]; }
#pragma unroll
  for (int o = 16; o > 0; o >>= 1) {
    s1 += __shfl_down(s1, o);
    s2 += __shfl_down(s2, o);
  }
  __shared__ double r1[8];
  __shared__ double r2[8];
  if ((threadIdx.x & 31) == 0) { r1[threadIdx.x >> 5] = s1; r2[threadIdx.x >> 5] = s2; }
  __syncthreads();
  if (threadIdx.x == 0) {
    double t1 = 0.0, t2 = 0.0;
#pragma unroll
    for (int i = 0; i < 8; ++i) { t1 += r1[i]; t2 += r2[i]; }
    out[0] = (float)(t1 / t2);
  }
}

// ---------------- launcher ----------------
extern "C" void kernel_launch(void* const* d_in, const int* in_sizes, int n_in,
                              void* d_out, int out_size, void* d_ws, size_t ws_size,
                              hipStream_t stream) {
  (void)in_sizes; (void)n_in; (void)out_size; (void)ws_size;

  const float* fake  = (const float*)d_in[0];
  const float* gamma = (const float*)d_in[1];
  const float* hdr   = (const float*)d_in[2];
  const float* rw    = (const float*)d_in[3];
  const float* ff    = (const float*)d_in[4];
  const float* gray  = (const float*)d_in[5];

  // ws layout: [0,16) gray-max bits (4 x u32); [64, 64+4096*8) part1; then part2
  unsigned int* gmax = (unsigned int*)d_ws;
  double* part1 = (double*)((char*)d_ws + 64);
  double* part2 = part1 + NBLOCKS_MAIN;

  il_init_ws<<<1, 64, 0, stream>>>(gmax);
  il_gray_max<<<dim3(256, NBATCH), 256, 0, stream>>>(gray, gmax);
  il_main<<<dim3(NTILES, NTILES, NBATCH), dim3(TS, TS), 0, stream>>>(
      fake, gamma, hdr, rw, ff, gmax, part1, part2);
  il_finalize<<<1, 256, 0, stream>>>(part1, part2, (float*)d_out, NBLOCKS_MAIN);
}